// SupervisedLoss_46377056862311
// MI455X (gfx1250) — compile-verified
//
#include <hip/hip_runtime.h>
#include <math.h>

// Problem constants (B=4, S=8, H=384, W=512 from setup_inputs)
#define NIMG 32
#define Hh   384
#define Ww   512
#define HW   (Hh*Ww)
#define NPIX (NIMG*HW)
#define NB   1024
#define EPSF 1e-6f
#define LMIN (-9.210340371976182f)   /* ln(1e-4) */
#define LMAX ( 9.210340371976182f)   /* ln(1e4)  */

typedef __attribute__((ext_vector_type(2))) float v2f;
typedef __attribute__((ext_vector_type(8))) float v8f;

// ---------------- workspace layout (bytes) ----------------
static constexpr size_t OFF_H1 = 0;                                  // u32[NIMG*NB]
static constexpr size_t OFF_H2 = OFF_H1 + (size_t)NIMG*NB*4;         // u32[NIMG*NB]
static constexpr size_t OFF_MS = OFF_H2 + (size_t)NIMG*NB*4;         // u32[NIMG] msum
static constexpr size_t OFF_FC = OFF_MS + (size_t)NIMG*4;            // u32[NIMG] fcnt
static constexpr size_t OFF_LO = OFF_FC + (size_t)NIMG*4;            // f32[NIMG]
static constexpr size_t OFF_HI = OFF_LO + (size_t)NIMG*4;            // f32[NIMG]
static constexpr size_t OFF_CB = OFF_HI + (size_t)NIMG*4;            // u32[NIMG] cum-before
static constexpr size_t OFF_TG = OFF_CB + (size_t)NIMG*4;            // u32[NIMG] target rank
static constexpr size_t OFF_SC = OFF_TG + (size_t)NIMG*4;            // f32[NIMG] scale
static constexpr size_t OFF_AC = OFF_SC + (size_t)NIMG*4;            // f32[8] accumulators
static constexpr size_t WS_NEEDED = OFF_AC + 8*4;
// accum: 0=depth_num 1=vm_sum 2=points_num 3=normal_num 4=normal_den 5=pose_sum

// ---------------- pass 1: coarse log histogram ----------------
__global__ void k_hist1(const float* __restrict__ dp, const float* __restrict__ dg,
                        const unsigned char* __restrict__ msk,
                        unsigned* __restrict__ hist1, unsigned* __restrict__ msum,
                        unsigned* __restrict__ fcnt)
{
  __shared__ unsigned lh[NB];
  __shared__ unsigned lms, lfc;
  const int img = blockIdx.x;
  for (int t = threadIdx.x; t < NB; t += 256) lh[t] = 0u;
  if (threadIdx.x == 0) { lms = 0u; lfc = 0u; }
  __syncthreads();

  const size_t base = (size_t)img * HW;
  unsigned my_m = 0, my_f = 0;
  const float binscale = (float)NB / (LMAX - LMIN);
  for (int p = threadIdx.x; p < HW; p += 256) {
    __builtin_prefetch(dp + base + p + 2048, 0, 0);   // global_prefetch_b8
    const float a = dp[base + p];
    const float g = dg[base + p];
    const bool  mm = (msk[base + p] != 0) && __builtin_isfinite(a) && (a > EPSF);
    const float r  = a / fmaxf(g, EPSF);
    const bool  fm = mm && __builtin_isfinite(r);
    my_m += mm ? 1u : 0u;
    my_f += fm ? 1u : 0u;
    if (fm) {
      int b = (int)((logf(r) - LMIN) * binscale);
      b = min(max(b, 0), NB - 1);
      atomicAdd(&lh[b], 1u);
    }
  }
  atomicAdd(&lms, my_m);
  atomicAdd(&lfc, my_f);
  __syncthreads();
  for (int t = threadIdx.x; t < NB; t += 256) hist1[img*NB + t] = lh[t];
  if (threadIdx.x == 0) { msum[img] = lms; fcnt[img] = lfc; }
}

// ---------------- select coarse bin containing the median ----------------
__global__ void k_sel1(const unsigned* __restrict__ hist1, const unsigned* __restrict__ fcnt,
                       unsigned* __restrict__ cbefore, unsigned* __restrict__ target,
                       float* __restrict__ loA, float* __restrict__ hiA)
{
  const int img = threadIdx.x;
  if (img >= NIMG) return;
  const unsigned fc  = fcnt[img];
  const unsigned idx = fc ? ((fc - 1u) >> 1) : 0u;   // max((fcnt-1)//2, 0)
  const unsigned tgt = idx + 1u;                     // rank (1-based)
  unsigned cum = 0, cb = 0;
  int sel = -1;
  for (int b = 0; b < NB; ++b) {
    const unsigned c = hist1[img*NB + b];
    if (sel < 0 && cum + c >= tgt) { sel = b; cb = cum; }
    cum += c;
  }
  if (sel < 0) { sel = NB - 1; cb = 0; }
  target[img]  = tgt;
  cbefore[img] = cb;
  const float wdt = (LMAX - LMIN) / (float)NB;
  loA[img] = expf(LMIN + sel * wdt);
  hiA[img] = expf(LMIN + (sel + 1) * wdt);
}

// ---------------- pass 2: fine linear histogram inside chosen bin ----------------
__global__ void k_hist2(const float* __restrict__ dp, const float* __restrict__ dg,
                        const unsigned char* __restrict__ msk,
                        const float* __restrict__ loA, const float* __restrict__ hiA,
                        unsigned* __restrict__ hist2)
{
  __shared__ unsigned lh[NB];
  const int img = blockIdx.x;
  for (int t = threadIdx.x; t < NB; t += 256) lh[t] = 0u;
  __syncthreads();

  const float lo = loA[img], hi = hiA[img];
  const float invw = (float)NB / fmaxf(hi - lo, 1e-30f);
  const size_t base = (size_t)img * HW;
  for (int p = threadIdx.x; p < HW; p += 256) {
    const float a = dp[base + p];
    const float g = dg[base + p];
    const bool  mm = (msk[base + p] != 0) && __builtin_isfinite(a) && (a > EPSF);
    const float r  = a / fmaxf(g, EPSF);
    const bool  fm = mm && __builtin_isfinite(r);
    if (fm && r >= lo && r < hi) {
      int b = (int)((r - lo) * invw);
      b = min(max(b, 0), NB - 1);
      atomicAdd(&lh[b], 1u);
    }
  }
  __syncthreads();
  for (int t = threadIdx.x; t < NB; t += 256) hist2[img*NB + t] = lh[t];
}

// ---------------- finish median -> per-image scale ----------------
__global__ void k_sel2(const unsigned* __restrict__ hist2, const unsigned* __restrict__ msum,
                       const unsigned* __restrict__ fcnt, const unsigned* __restrict__ cbefore,
                       const unsigned* __restrict__ target, const float* __restrict__ loA,
                       const float* __restrict__ hiA, float* __restrict__ scaleArr)
{
  const int img = threadIdx.x;
  if (img >= NIMG) return;
  const unsigned tgt = target[img];
  unsigned cum = cbefore[img];
  int sel = -1;
  for (int b = 0; b < NB; ++b) {
    const unsigned c = hist2[img*NB + b];
    if (sel < 0 && cum + c >= tgt) sel = b;
    cum += c;
  }
  if (sel < 0) sel = NB - 1;
  const float lo = loA[img], hi = hiA[img];
  const float med = lo + (sel + 0.5f) * (hi - lo) / (float)NB;
  float sc = fminf(fmaxf(med, 0.001f), 1000.0f);
  if (msum[img] < 16u || fcnt[img] == 0u) sc = 1.0f;
  scaleArr[img] = sc;
}

// ---------------- pose loss (mat_to_quat + pose_encoding + smooth L1) ----------------
__global__ void k_pose(const float* __restrict__ posePred, const float* __restrict__ Kall,
                       const float* __restrict__ Pose, const float* __restrict__ scaleArr,
                       float* __restrict__ accum)
{
  const int i = threadIdx.x;
  if (i >= NIMG) return;
  const float* M = Pose + i*16;
  const float* K = Kall + i*9;
  const float sc = scaleArr[i];
  const float m00=M[0], m01=M[1], m02=M[2],  tx=M[3];
  const float m10=M[4], m11=M[5], m12=M[6],  ty=M[7];
  const float m20=M[8], m21=M[9], m22=M[10], tz=M[11];

  float qa[4];
  qa[0] = sqrtf(fmaxf(1.0f + m00 + m11 + m22, 0.0f));
  qa[1] = sqrtf(fmaxf(1.0f + m00 - m11 - m22, 0.0f));
  qa[2] = sqrtf(fmaxf(1.0f - m00 + m11 - m22, 0.0f));
  qa[3] = sqrtf(fmaxf(1.0f - m00 - m11 + m22, 0.0f));
  float cand[4][4] = {
    { qa[0]*qa[0], m21-m12,     m02-m20,     m10-m01     },
    { m21-m12,     qa[1]*qa[1], m10+m01,     m02+m20     },
    { m02-m20,     m10+m01,     qa[2]*qa[2], m12+m21     },
    { m10-m01,     m20+m02,     m21+m12,     qa[3]*qa[3] }
  };
  int best = 0; float bv = qa[0];
  for (int j = 1; j < 4; ++j) if (qa[j] > bv) { bv = qa[j]; best = j; }
  const float inv2 = 1.0f / (2.0f * fmaxf(qa[best], 0.1f));
  float q0 = cand[best][0]*inv2, q1 = cand[best][1]*inv2,
        q2 = cand[best][2]*inv2, q3 = cand[best][3]*inv2;
  if (q0 < 0.0f) { q0=-q0; q1=-q1; q2=-q2; q3=-q3; }

  const float fovh = 2.0f * atanf(((float)Hh * 0.5f) / K[4]);
  const float fovw = 2.0f * atanf(((float)Ww * 0.5f) / K[0]);
  const float pg[9] = { tx*sc, ty*sc, tz*sc, q0, q1, q2, q3, fovh, fovw };

  float local = 0.0f;
  for (int j = 0; j < 9; ++j) {
    const float d = fabsf(posePred[i*9 + j] - pg[j]);
    local += (d < 1.0f) ? 0.5f*d*d : d - 0.5f;
  }
  atomicAdd(&accum[5], local);
}

// ---------------- helpers for main pass ----------------
__device__ __forceinline__ void camPoint(const float* __restrict__ d, size_t ibase,
                                         int h, int w, float fx, float fy, float cx, float cy,
                                         float& x, float& y, float& z)
{
  const float dd = d[ibase + (size_t)h*Ww + w];
  x = ((float)w - cx) / fx * dd;
  y = ((float)h - cy) / fy * dd;
  z = dd;
}

__device__ __forceinline__ void normalAt(const float* __restrict__ dep, size_t ibase,
                                         int h, int w, float fx, float fy, float cx, float cy,
                                         float& nx, float& ny, float& nz)
{
  const int hm = max(h-1, 0), hp = min(h+1, Hh-1);
  const int wm = max(w-1, 0), wp = min(w+1, Ww-1);
  float ax,ay,az, bx,by,bz, ux,uy,uz, vx,vy,vz;
  camPoint(dep, ibase, h,  wp, fx,fy,cx,cy, ax,ay,az);
  camPoint(dep, ibase, h,  wm, fx,fy,cx,cy, bx,by,bz);
  camPoint(dep, ibase, hp, w,  fx,fy,cx,cy, ux,uy,uz);
  camPoint(dep, ibase, hm, w,  fx,fy,cx,cy, vx,vy,vz);
  const float fxdx = ax-bx, fxdy = ay-by, fxdz = az-bz;   // x-direction diff
  const float fydx = ux-vx, fydy = uy-vy, fydz = uz-vz;   // y-direction diff
  // cross(fy_diff, fx_diff)
  const float cxv = fydy*fxdz - fydz*fxdy;
  const float cyv = fydz*fxdx - fydx*fxdz;
  const float czv = fydx*fxdy - fydy*fxdx;
  const float n   = sqrtf(cxv*cxv + cyv*cyv + czv*czv);
  const float inv = 1.0f / fmaxf(n, EPSF);
  nx = cxv*inv; ny = cyv*inv; nz = czv*inv;
}

// ---------------- main streaming pass ----------------
// World-point transform via V_WMMA_F32_16X16X4_F32, shuffle-free:
//   A layout (ISA 32-bit 16x4): lanes 0-15 hold A[m][0..1], lanes 16-31 hold A[m][2..3], m=lane&15.
//   B layout (K split):         v0 = K0 (lanes 0-15) / K2 (lanes 16-31); v1 = K1 / K3.
//   D layout:                   lane n      VGPR r = D[r][n]     (rows 0-7)
//                               lane 16+n   VGPR r = D[8+r][n]   (rows 8-15)
// WMMA#1 puts [R|t] in A-rows 0-2, B-columns = points of lanes 0-15
//   -> lane n (<16) reads its world point from C1[0..2].
// WMMA#2 puts [R|t] in A-rows 8-10, B-columns = points of lanes 16-31
//   -> lane 16+n reads its world point from C2[0..2]  (rows 8-10 live in the hi half-lanes).
// img/h/w0 are computed from blockIdx only (block = 256 consecutive pixels of one row),
// so pose/intrinsics/scale loads are wave-uniform scalar loads and the per-lane A/B
// operands are pure v_cndmask selects on scalar sources: EXEC stays all-ones throughout
// (WMMA requirement) and there is no divergent control flow in the hot path.
__global__ void k_main(const float* __restrict__ dp, const float* __restrict__ pp,
                       const float* __restrict__ dg, const float* __restrict__ Kall,
                       const float* __restrict__ Pose, const unsigned char* __restrict__ msk,
                       const float* __restrict__ scaleArr, float* __restrict__ accum)
{
  const int tid  = threadIdx.x;
  const int lane = tid & 31;

  // block-uniform coordinates (HW % 256 == 0, Ww % 256 == 0)
  const int blk  = blockIdx.x;
  const int img  = (blk * 256) / HW;          // uniform
  const int off0 = (blk * 256) % HW;          // uniform
  const int h    = off0 / Ww;                 // uniform
  const int w0   = off0 % Ww;                 // uniform
  const int w    = w0 + tid;
  const size_t pix = (size_t)img * HW + (size_t)off0 + (size_t)tid;

  // uniform scalar parameters
  const float* K = Kall + img*9;
  const float fx = fmaxf(K[0], EPSF), fy = fmaxf(K[4], EPSF);
  const float cx = K[2], cy = K[5];
  const float sc = scaleArr[img];
  const float* M = Pose + img*16;
  const float R00=M[0], R01=M[1], R02=M[2],  T0=M[3];
  const float R10=M[4], R11=M[5], R12=M[6],  T1=M[7];
  const float R20=M[8], R21=M[9], R22=M[10], T2=M[11];

  const float dpv = dp[pix];
  const float dgv = dg[pix];
  const bool  vm  = (msk[pix] != 0) && __builtin_isfinite(dgv) && (dgv > 1e-6f);
  const float vmf = vm ? 1.0f : 0.0f;
  const float dga = dgv * sc;

  const float depth_num = fabsf(dpv - dga) * vmf;

  // own camera-space point from aligned gt depth (sanitized)
  const float dpt = vm ? dga : 0.0f;
  const float px = ((float)w - cx) / fx * dpt;
  const float py = ((float)h - cy) / fy * dpt;
  const float pz = dpt;

  // partner (lane^16) pixel: reload + recompute locally (no cross-lane ops)
  const size_t pix2 = pix ^ 16u;
  const int    w2   = w ^ 16;
  const float  dgv2 = dg[pix2];
  const bool   vm2  = (msk[pix2] != 0) && __builtin_isfinite(dgv2) && (dgv2 > 1e-6f);
  const float  dpt2 = vm2 ? dgv2 * sc : 0.0f;
  const float  px2  = ((float)w2 - cx) / fx * dpt2;
  const float  py2  = ((float)h  - cy) / fy * dpt2;
  const float  pz2  = dpt2;

  const int  m      = lane & 15;
  const bool lohalf = lane < 16;

  // WMMA #1: [R|t] in A-rows 0..2; B columns = points of lanes 0..15
  // selects on scalar-register sources -> v_cndmask chains, no loads, no branches
  const float a0 = (m == 0) ? (lohalf ? R00 : R02)
                 : (m == 1) ? (lohalf ? R10 : R12)
                 : (m == 2) ? (lohalf ? R20 : R22) : 0.0f;
  const float a1 = (m == 0) ? (lohalf ? R01 : T0)
                 : (m == 1) ? (lohalf ? R11 : T1)
                 : (m == 2) ? (lohalf ? R21 : T2) : 0.0f;
  v2f A1; A1.x = a0; A1.y = a1;
  v2f B1; B1.x = lohalf ? px : pz2;       // hi lane 16+n supplies K2 = z of point n (=partner)
          B1.y = lohalf ? py : 1.0f;      // hi lane supplies K3 = homogeneous 1
  v8f C1 = {};
  C1 = __builtin_amdgcn_wmma_f32_16x16x4_f32(false, A1, false, B1, (short)0, C1, false, false);

  // WMMA #2: [R|t] in A-rows 8..10; B columns = points of lanes 16..31
  const float c0 = (m == 8)  ? (lohalf ? R00 : R02)
                 : (m == 9)  ? (lohalf ? R10 : R12)
                 : (m == 10) ? (lohalf ? R20 : R22) : 0.0f;
  const float c1 = (m == 8)  ? (lohalf ? R01 : T0)
                 : (m == 9)  ? (lohalf ? R11 : T1)
                 : (m == 10) ? (lohalf ? R21 : T2) : 0.0f;
  v2f A2; A2.x = c0; A2.y = c1;
  v2f B2; B2.x = lohalf ? px2 : pz;       // lo lane n supplies K0/K1 = x,y of point 16+n (=partner)
          B2.y = lohalf ? py2 : 1.0f;     // hi lane 16+n supplies own z and 1
  v8f C2 = {};
  C2 = __builtin_amdgcn_wmma_f32_16x16x4_f32(false, A2, false, B2, (short)0, C2, false, false);

  // every lane now owns its world point without any cross-lane move
  const float wx = lohalf ? C1[0] : C2[0];
  const float wy = lohalf ? C1[1] : C2[1];
  const float wz = lohalf ? C1[2] : C2[2];

  const float* q = pp + pix*3;
  const float points_num = (fabsf(q[0]-wx) + fabsf(q[1]-wy) + fabsf(q[2]-wz)) * vmf;

  // normals (interior only); pred from depth_pred, gt from RAW depth_gt (per reference)
  const bool interior = (h > 0) && (h < Hh-1) && (w > 0) && (w < Ww-1);
  const float nmf = (vm && interior) ? 1.0f : 0.0f;
  float npx,npy,npz, ngx,ngy,ngz;
  normalAt(dp, (size_t)img*HW, h, w, fx,fy,cx,cy, npx,npy,npz);
  normalAt(dg, (size_t)img*HW, h, w, fx,fy,cx,cy, ngx,ngy,ngz);
  // masked_cosine_loss re-normalizes both
  {
    const float lp = sqrtf(npx*npx+npy*npy+npz*npz); const float il = 1.0f/fmaxf(lp, EPSF);
    npx*=il; npy*=il; npz*=il;
    const float lg = sqrtf(ngx*ngx+ngy*ngy+ngz*ngz); const float ig = 1.0f/fmaxf(lg, EPSF);
    ngx*=ig; ngy*=ig; ngz*=ig;
  }
  const float cosv = fminf(fmaxf(npx*ngx + npy*ngy + npz*ngz, -1.0f), 1.0f);
  const float normal_num = (1.0f - cosv) * nmf;

  // block reduction of 5 accumulators, then one global atomic each
  float vals[5] = { depth_num, vmf, points_num, normal_num, nmf };
  __shared__ float sac[5];
  if (tid < 5) sac[tid] = 0.0f;
  __syncthreads();
#pragma unroll
  for (int k = 0; k < 5; ++k) {
    float v = vals[k];
    for (int o = 16; o > 0; o >>= 1) v += __shfl_down(v, o);
    if (lane == 0) atomicAdd(&sac[k], v);
  }
  __syncthreads();
  if (tid < 5) atomicAdd(&accum[tid], sac[tid]);
}

// ---------------- finalize ----------------
__global__ void k_final(const float* __restrict__ accum, float* __restrict__ out)
{
  const float pose_l   = accum[5] * (1.0f / 288.0f);                 // mean over B*S*9
  const float depth_l  = accum[0] / fmaxf(accum[1], 1.0f);
  const float points_l = accum[2] / fmaxf(3.0f * accum[1], 1.0f);    // mask broadcast x3
  const float normal_l = accum[3] / fmaxf(accum[4], 1.0f);
  out[0] = 1.0f*pose_l + 1.0f*depth_l + 1.0f*points_l + 0.1f*normal_l;
}

// ---------------- launcher ----------------
extern "C" void kernel_launch(void* const* d_in, const int* in_sizes, int n_in,
                              void* d_out, int out_size, void* d_ws, size_t ws_size,
                              hipStream_t stream)
{
  const float* depth_pred = (const float*)d_in[0];
  const float* points_pred= (const float*)d_in[1];
  const float* pose_pred  = (const float*)d_in[2];
  const float* depth_gt   = (const float*)d_in[3];
  const float* intrinsics = (const float*)d_in[4];
  const float* pose_gt    = (const float*)d_in[5];
  const unsigned char* valid_mask = (const unsigned char*)d_in[6];
  float* out = (float*)d_out;

  char* ws = (char*)d_ws;
  unsigned* hist1   = (unsigned*)(ws + OFF_H1);
  unsigned* hist2   = (unsigned*)(ws + OFF_H2);
  unsigned* msum    = (unsigned*)(ws + OFF_MS);
  unsigned* fcnt    = (unsigned*)(ws + OFF_FC);
  float*    loA     = (float*)   (ws + OFF_LO);
  float*    hiA     = (float*)   (ws + OFF_HI);
  unsigned* cbefore = (unsigned*)(ws + OFF_CB);
  unsigned* target  = (unsigned*)(ws + OFF_TG);
  float*    scaleA  = (float*)   (ws + OFF_SC);
  float*    accum   = (float*)   (ws + OFF_AC);

  hipMemsetAsync(d_ws, 0, WS_NEEDED, stream);

  k_hist1<<<NIMG, 256, 0, stream>>>(depth_pred, depth_gt, valid_mask, hist1, msum, fcnt);
  k_sel1 <<<1, 32, 0, stream>>>(hist1, fcnt, cbefore, target, loA, hiA);
  k_hist2<<<NIMG, 256, 0, stream>>>(depth_pred, depth_gt, valid_mask, loA, hiA, hist2);
  k_sel2 <<<1, 32, 0, stream>>>(hist2, msum, fcnt, cbefore, target, loA, hiA, scaleA);
  k_pose <<<1, 32, 0, stream>>>(pose_pred, intrinsics, pose_gt, scaleA, accum);
  k_main <<<NPIX/256, 256, 0, stream>>>(depth_pred, points_pred, depth_gt, intrinsics,
                                        pose_gt, valid_mask, scaleA, accum);
  k_final<<<1, 1, 0, stream>>>(accum, out);
}